// SRlayer_80805514707487
// MI455X (gfx1250) — compile-verified
//
#include <hip/hip_runtime.h>
#include <hip/hip_bf16.h>
#include <cstddef>

// Problem constants (from reference setup_inputs)
#define SS   512
#define BB   16
#define HH   16
#define DD   64
#define NIN  1024   // IN = H*D

typedef __attribute__((ext_vector_type(2))) float v2f;
typedef __attribute__((ext_vector_type(8))) float v8f;

// ---------------------------------------------------------------------------
// Kernel 0: transpose + pair-interleave out_w (IN, H*D) into
//   wP[((k>>1)*NIN + n)*2 + (k&1)] = out_w[n*NIN + k]
// so each GEMM B-fragment (two consecutive k for one n) is a single
// 8-byte-aligned global_load_b64, coalesced across lanes.
// ---------------------------------------------------------------------------
__global__ __launch_bounds__(256) void srwm_pack_kernel(
    const float* __restrict__ src, float* __restrict__ dst)
{
    __shared__ float tile[32][33];
    const int bx = blockIdx.x, by = blockIdx.y;
    const int tx = threadIdx.x & 31, ty = threadIdx.x >> 5;   // 32 x 8
    #pragma unroll
    for (int i = 0; i < 4; ++i)
        tile[ty + 8 * i][tx] = src[(size_t)(by * 32 + ty + 8 * i) * NIN + bx * 32 + tx];
    __syncthreads();
    #pragma unroll
    for (int i = 0; i < 4; ++i) {
        const int k = bx * 32 + ty + 8 * i;   // column of src = row of wT
        const int n = by * 32 + tx;
        dst[((size_t)(k >> 1) * NIN + n) * 2 + (k & 1)] = tile[tx][ty + 8 * i];
    }
}

// ---------------------------------------------------------------------------
// Kernel 1: sequential SRWM scan. One block per (b,h) chain; W state lives in
// registers (thread (row,quad) owns W[row][quad*16 .. quad*16+15]); vectors
// broadcast via LDS; dot products reduced with wave32 shfl_xor butterflies.
// Writes y_t into `ys` (staged in the `out` region of d_out) and the final
// state deltas into dWy/dWq/dWk/dwb.
// ---------------------------------------------------------------------------
__global__ __launch_bounds__(256, 1) void srwm_scan_kernel(
    const float* __restrict__ hin,                         // (S,B,IN)
    const float* __restrict__ sWy, const float* __restrict__ sWq,
    const float* __restrict__ sWk, const float* __restrict__ swb,
    const float* __restrict__ Wy0, const float* __restrict__ Wq0,
    const float* __restrict__ Wk0, const float* __restrict__ wb0,
    float* __restrict__ ys,                                // (S,B,IN)
    float* __restrict__ dWy, float* __restrict__ dWq,
    float* __restrict__ dWk, float* __restrict__ dwb)
{
    const int bid  = blockIdx.x;        // b*16 + h
    const int b    = bid >> 4;
    const int hh   = bid & 15;
    const int tid  = threadIdx.x;       // 256 threads = 8 wave32
    const int row  = tid >> 2;          // 0..63
    const int quad = tid & 3;           // 0..3 (16 cols each)
    const int c0   = quad << 4;
    const int lane = tid & 31;
    const int wav  = tid >> 5;

    __shared__ float xv[64];            // softmax(x_t)
    __shared__ float kv[64];            // softmax(k)
    __shared__ float dq[64];            // q - k
    __shared__ float xraw[64];
    __shared__ float wredA[8], wredB[8];
    __shared__ float bpart[32];         // [wave][quad] partials
    __shared__ float bsig[4];           // sigmoid(beta)
    __shared__ float dvbS[4];           // wb^T (q-k)

    // --- load initial carry: state + broadcast init weights -----------------
    float wy[16], wq[16], wk[16], wbv;
    const size_t Woff  = (((size_t)bid << 6) + row) * 64 + c0;  // (b,h,row, c0)
    const size_t W0off = (((size_t)hh  << 6) + row) * 64 + c0;  // (0,h,row, c0)
    #pragma unroll
    for (int j = 0; j < 16; ++j) {
        wy[j] = sWy[Woff + j] + Wy0[W0off + j];
        wq[j] = sWq[Woff + j] + Wq0[W0off + j];
        wk[j] = sWk[Woff + j] + Wk0[W0off + j];
    }
    wbv = swb[(((size_t)bid << 6) + row) * 4 + quad] +
          wb0[(((size_t)hh  << 6) + row) * 4 + quad];

    for (int t = 0; t < SS; ++t) {
        // ---- x_t = softmax(h[t,b,h*64 : +64]) ------------------------------
        if (tid < 64)
            xraw[tid] = hin[(((size_t)t * BB + b) << 10) + (hh << 6) + tid];
        __syncthreads();
        float xr = xraw[row];                       // all 4 quads hold row val
        float mx = xr;
        #pragma unroll
        for (int m = 1; m < 32; m <<= 1) mx = fmaxf(mx, __shfl_xor(mx, m, 32));
        if (lane == 0) wredA[wav] = mx;
        __syncthreads();
        mx = wredA[0];
        #pragma unroll
        for (int i = 1; i < 8; ++i) mx = fmaxf(mx, wredA[i]);
        float ex = __expf(xr - mx);
        float ps = (quad == 0) ? ex : 0.f;          // count each row once
        #pragma unroll
        for (int m = 1; m < 32; m <<= 1) ps += __shfl_xor(ps, m, 32);
        __syncthreads();                            // wredB safe to rewrite
        if (lane == 0) wredB[wav] = ps;
        __syncthreads();
        float ssum = 0.f;
        #pragma unroll
        for (int i = 0; i < 8; ++i) ssum += wredB[i];
        if (quad == 0) xv[row] = ex / ssum;
        __syncthreads();

        // ---- pass 1: y = Wy x, qpre = Wq x, kpre = Wk x, beta = wb^T x -----
        float py = 0.f, pq = 0.f, pk = 0.f;
        #pragma unroll
        for (int j = 0; j < 16; ++j) {
            const float xj = xv[c0 + j];
            py = fmaf(wy[j], xj, py);
            pq = fmaf(wq[j], xj, pq);
            pk = fmaf(wk[j], xj, pk);
        }
        py += __shfl_xor(py, 1, 32); py += __shfl_xor(py, 2, 32);
        pq += __shfl_xor(pq, 1, 32); pq += __shfl_xor(pq, 2, 32);
        pk += __shfl_xor(pk, 1, 32); pk += __shfl_xor(pk, 2, 32);
        if (quad == 0)                              // emit y_t
            ys[(((size_t)t * BB + b) << 10) + (hh << 6) + row] = py;

        float pb = wbv * xv[row];                   // beta partial (row,quad)
        pb += __shfl_xor(pb, 4, 32);
        pb += __shfl_xor(pb, 8, 32);
        pb += __shfl_xor(pb, 16, 32);
        if (lane < 4) bpart[(wav << 2) | lane] = pb;

        // ---- joint softmax of q and k --------------------------------------
        float mq = pq, mk = pk;
        #pragma unroll
        for (int m = 1; m < 32; m <<= 1) {
            mq = fmaxf(mq, __shfl_xor(mq, m, 32));
            mk = fmaxf(mk, __shfl_xor(mk, m, 32));
        }
        __syncthreads();                            // wredA/B rewrite safe
        if (lane == 0) { wredA[wav] = mq; wredB[wav] = mk; }
        __syncthreads();
        mq = wredA[0]; mk = wredB[0];
        #pragma unroll
        for (int i = 1; i < 8; ++i) {
            mq = fmaxf(mq, wredA[i]); mk = fmaxf(mk, wredB[i]);
        }
        float eq = __expf(pq - mq), ek = __expf(pk - mk);
        float sqp = (quad == 0) ? eq : 0.f, skp = (quad == 0) ? ek : 0.f;
        #pragma unroll
        for (int m = 1; m < 32; m <<= 1) {
            sqp += __shfl_xor(sqp, m, 32);
            skp += __shfl_xor(skp, m, 32);
        }
        __syncthreads();
        if (lane == 0) { wredA[wav] = sqp; wredB[wav] = skp; }
        __syncthreads();
        float sq = 0.f, sk = 0.f;
        #pragma unroll
        for (int i = 0; i < 8; ++i) { sq += wredA[i]; sk += wredB[i]; }
        const float qn = eq / sq, kn = ek / sk;
        if (quad == 0) { kv[row] = kn; dq[row] = qn - kn; }
        if (tid < 4) {                              // beta = sigmoid(sum)
            float s = 0.f;
            #pragma unroll
            for (int w = 0; w < 8; ++w) s += bpart[(w << 2) | tid];
            bsig[tid] = 1.f / (1.f + __expf(-s));
        }
        __syncthreads();

        // ---- pass 2: dv = W (q-k), dvb = wb^T (q-k) ------------------------
        float pdy = 0.f, pdq = 0.f, pdk = 0.f;
        #pragma unroll
        for (int j = 0; j < 16; ++j) {
            const float dj = dq[c0 + j];
            pdy = fmaf(wy[j], dj, pdy);
            pdq = fmaf(wq[j], dj, pdq);
            pdk = fmaf(wk[j], dj, pdk);
        }
        pdy += __shfl_xor(pdy, 1, 32); pdy += __shfl_xor(pdy, 2, 32);
        pdq += __shfl_xor(pdq, 1, 32); pdq += __shfl_xor(pdq, 2, 32);
        pdk += __shfl_xor(pdk, 1, 32); pdk += __shfl_xor(pdk, 2, 32);

        float pdb = wbv * dq[row];
        pdb += __shfl_xor(pdb, 4, 32);
        pdb += __shfl_xor(pdb, 8, 32);
        pdb += __shfl_xor(pdb, 16, 32);
        if (lane < 4) bpart[(wav << 2) | lane] = pdb;
        __syncthreads();
        if (tid < 4) {
            float s = 0.f;
            #pragma unroll
            for (int w = 0; w < 8; ++w) s += bpart[(w << 2) | tid];
            dvbS[tid] = s;
        }
        __syncthreads();

        // ---- rank-1 updates (pre-update W used above) ----------------------
        const float a0 = bsig[0] * pdy;
        const float a1 = bsig[1] * pdq;
        const float a2 = bsig[2] * pdk;
        #pragma unroll
        for (int j = 0; j < 16; ++j) {
            const float kj = kv[c0 + j];
            wy[j] = fmaf(a0, kj, wy[j]);
            wq[j] = fmaf(a1, kj, wq[j]);
            wk[j] = fmaf(a2, kj, wk[j]);
        }
        wbv = fmaf(bsig[3] * kv[row], dvbS[quad], wbv);
        // next iteration's first __syncthreads() fences LDS reuse
    }

    // ---- final state deltas: W_final - W_init ------------------------------
    #pragma unroll
    for (int j = 0; j < 16; ++j) {
        dWy[Woff + j] = wy[j] - Wy0[W0off + j];
        dWq[Woff + j] = wq[j] - Wq0[W0off + j];
        dWk[Woff + j] = wk[j] - Wk0[W0off + j];
    }
    dwb[(((size_t)bid << 6) + row) * 4 + quad] =
        wbv - wb0[(((size_t)hh << 6) + row) * 4 + quad];
}

// ---------------------------------------------------------------------------
// Kernel 2: out = h + ys @ out_w^T using V_WMMA_F32_16X16X4_F32.
// ys is staged in `outp` and overwritten in place: each block owns rows
// m0..m0+15 exclusively (reads A only from those rows), __syncthreads()
// separates the last A read from the first store.
// Block = 256 threads = 8 waves; wave w computes the 16 x 128 tile at
// columns [w*128, w*128+128); K-loop over 1024 in steps of 4.
// A and B fragments are single global_load_b64 issues (9 loads : 8 WMMAs).
// ---------------------------------------------------------------------------
__global__ __launch_bounds__(256, 1) void srwm_out_gemm_kernel(
    const float* __restrict__ hin,   // (S*B, 1024) residual input
    const float* __restrict__ wP,    // pair-interleaved out_w^T (see pack)
    float* __restrict__ outp)        // in: ys ; out: h + ys @ out_w^T
{
    const int m0   = blockIdx.x << 4;           // 16 rows per block
    const int wav  = threadIdx.x >> 5;
    const int lane = threadIdx.x & 31;
    const int n0   = wav << 7;                  // 128 cols per wave
    const int nl   = lane & 15;                 // M index (A) / N index (B,D)
    const int kk   = (lane >> 4) << 1;          // K sub-offset: 0 or 2

    v8f acc[8];
    #pragma unroll
    for (int s = 0; s < 8; ++s)
        #pragma unroll
        for (int r = 0; r < 8; ++r) acc[s][r] = 0.f;

    const float* __restrict__ ysm = outp + (size_t)m0 * NIN;

    for (int k0 = 0; k0 < NIN; k0 += 4) {
        // A fragment: 16x4 tile of ys rows m0..m0+15 (one b64 per lane)
        const v2f a = *reinterpret_cast<const v2f*>(
            ysm + (size_t)nl * NIN + k0 + kk);
        const float* __restrict__ wrow = wP + ((size_t)((k0 + kk) >> 1) * NIN) * 2;
        #pragma unroll
        for (int s = 0; s < 8; ++s) {
            // B fragment: 4x16 tile, pair-interleaved -> one b64 per lane
            const int n = n0 + (s << 4) + nl;
            const v2f bf = *reinterpret_cast<const v2f*>(wrow + 2 * n);
            acc[s] = __builtin_amdgcn_wmma_f32_16x16x4_f32(
                false, a, false, bf, (short)0, acc[s], false, false);
        }
    }

    __syncthreads();   // all ys reads in this block done before in-place write

    #pragma unroll
    for (int s = 0; s < 8; ++s) {
        #pragma unroll
        for (int r = 0; r < 8; ++r) {
            const int mm = m0 + ((lane >> 4) << 3) + r;   // D layout: M=(l>>4)*8+r
            const int nn = n0 + (s << 4) + nl;            // N = l&15
            const size_t idx = (size_t)mm * NIN + nn;
            outp[idx] = hin[idx] + acc[s][r];
        }
    }
}

// ---------------------------------------------------------------------------
extern "C" void kernel_launch(void* const* d_in, const int* in_sizes, int n_in,
                              void* d_out, int out_size, void* d_ws, size_t ws_size,
                              hipStream_t stream)
{
    const float* h     = (const float*)d_in[0];   // (S,B,IN)
    const float* sWy   = (const float*)d_in[1];   // (B,H,D,D)
    const float* sWq   = (const float*)d_in[2];
    const float* sWk   = (const float*)d_in[3];
    const float* swb   = (const float*)d_in[4];   // (B,H,D,4)
    const float* Wy0   = (const float*)d_in[5];   // (1,H,D,D)
    const float* Wq0   = (const float*)d_in[6];
    const float* Wk0   = (const float*)d_in[7];
    const float* wb0   = (const float*)d_in[8];   // (1,H,D,4)
    const float* out_w = (const float*)d_in[9];   // (IN, H*D)

    float* out = (float*)d_out;                   // (S,B,IN) = 8,388,608 floats
    float* dWy = out + (size_t)SS * BB * NIN;     // +8388608
    float* dWq = dWy + (size_t)BB * HH * DD * DD; // +1048576
    float* dWk = dWq + (size_t)BB * HH * DD * DD;
    float* dwb = dWk + (size_t)BB * HH * DD * DD;

    float* wP = (float*)d_ws;                     // 1024*1024 floats = 4 MB

    // 0) transpose + pair-interleave out_w for b64-coalesced GEMM B-loads
    srwm_pack_kernel<<<dim3(32, 32), 256, 0, stream>>>(out_w, wP);

    // 1) sequential scan: one block per (b,h) chain; ys staged into `out`
    srwm_scan_kernel<<<BB * HH, 256, 0, stream>>>(
        h, sWy, sWq, sWk, swb, Wy0, Wq0, Wk0, wb0, out, dWy, dWq, dWk, dwb);

    // 2) WMMA epilogue GEMM, in place on `out`
    srwm_out_gemm_kernel<<<(SS * BB) / 16, 256, 0, stream>>>(h, wP, out);
}